// get_loss_85160611545325
// MI455X (gfx1250) — compile-verified
//
#include <hip/hip_runtime.h>
#include <math.h>

typedef __attribute__((ext_vector_type(2))) float v2f;
typedef __attribute__((ext_vector_type(4))) float v4f;
typedef __attribute__((ext_vector_type(8))) float v8f;

#define N_PTS  4096
#define M_SKEL 100
#define KNN    10
#define BIGF   1e30f

__global__ void zero_out_kernel(float* out) { out[0] = 0.0f; }

// One wave = one 16-row tile of queries. 8 waves per block, grid = B*N/128 = 128.
// score[i][j] = -2*p_i . p_j + |p_j|^2  = d2(i,j) - |p_i|^2  (order-preserving shift,
// and the shift cancels in |d1 - d2| for the voronoi term).
__global__ __launch_bounds__(256) void voronoi_kernel(const float* __restrict__ xyz,
                                                      const float* __restrict__ skel,
                                                      float* __restrict__ out) {
  __shared__ float tiles[8][2][256];     // per-wave double-buffered 16x16 score tile
  __shared__ float mergeV[8][32][KNN];
  __shared__ int   mergeI[8][32][KNN];

  const int lane = threadIdx.x & 31;
  const int wave = threadIdx.x >> 5;
  const int col  = lane & 15;
  const int hi   = lane >> 4;

  const int rowBase = blockIdx.x * 128 + wave * 16;   // flat row id in [0, B*N)
  const int b  = rowBase >> 12;                       // N = 4096
  const int i0 = rowBase & (N_PTS - 1);
  const int i  = i0 + col;                            // this lane's query row

  // Query point (first 3 of the 6 features). 24-byte stride -> 8B aligned loads.
  const float* qp = xyz + (size_t)(b * N_PTS + i) * 6;
  const float qx = qp[0], qy = qp[1], qz = qp[2];

  // A fragment (16x4 f32): lanes 0-15 hold (K0,K1)=(x,y), lanes 16-31 hold (K2,K3)=(z,1)
  v2f afrag;
  afrag.x = hi ? qz : qx;
  afrag.y = hi ? 1.0f : qy;

  float best[KNN];
  int   bidx[KNN];
#pragma unroll
  for (int k = 0; k < KNN; ++k) { best[k] = BIGF; bidx[k] = -1; }

  const float* batch_pts = xyz + (size_t)b * N_PTS * 6;

  for (int jt = 0; jt < N_PTS / 16; ++jt) {
    // B fragment (4x16 f32): column j = jt*16+col is (-2x,-2y,-2z,|p_j|^2)
    const int j = jt * 16 + col;
    const float* pp = batch_pts + (size_t)j * 6;
    const float px = pp[0], py = pp[1], pz = pp[2];
    const float nrm2 = px * px + py * py + pz * pz;
    v2f bfrag;
    bfrag.x = hi ? (-2.0f * pz) : (-2.0f * px);
    bfrag.y = hi ? nrm2         : (-2.0f * py);

    v8f c = {};
    c = __builtin_amdgcn_wmma_f32_16x16x4_f32(
        /*neg_a=*/false, afrag, /*neg_b=*/false, bfrag,
        /*c_mod=*/(short)0, c, /*reuse_a=*/false, /*reuse_b=*/false);

    // Spill tile to LDS: C/D layout -> element (M = v + 8*hi, N = col) in vgpr v.
    float* tbuf = &tiles[wave][jt & 1][0];
#pragma unroll
    for (int v = 0; v < 8; ++v) tbuf[(v + 8 * hi) * 16 + col] = c[v];
    __syncthreads();   // uniform trip count across all 8 waves

    // Each lane consumes 8 values of row r = lane&15 (half selected by hi).
    // 32B-aligned contiguous slice -> two ds_load_b128.
    const float* rp = &tiles[wave][jt & 1][col * 16 + 8 * hi];
    const v4f c0 = *(const v4f*)(rp);
    const v4f c1 = *(const v4f*)(rp + 4);

    // Common-case fast reject: one branch for all 8 candidates.
    const float m8 = fminf(fminf(fminf(c0.x, c0.y), fminf(c0.z, c0.w)),
                           fminf(fminf(c1.x, c1.y), fminf(c1.z, c1.w)));
    if (m8 < best[KNN - 1]) {
      float cand[8] = {c0.x, c0.y, c0.z, c0.w, c1.x, c1.y, c1.z, c1.w};
#pragma unroll
      for (int t = 0; t < 8; ++t) {
        const float v = cand[t];
        if (v < best[KNN - 1]) {
          best[KNN - 1] = v;
          bidx[KNN - 1] = jt * 16 + 8 * hi + t;
#pragma unroll
          for (int k = KNN - 1; k > 0; --k) {
            if (best[k] < best[k - 1]) {
              float tv = best[k]; best[k] = best[k - 1]; best[k - 1] = tv;
              int   ti = bidx[k]; bidx[k] = bidx[k - 1]; bidx[k - 1] = ti;
            }
          }
        }
      }
    }
  }

  // Publish per-lane sorted top-10, then lanes 0-15 merge the two half-row lists.
#pragma unroll
  for (int k = 0; k < KNN; ++k) { mergeV[wave][lane][k] = best[k]; mergeI[wave][lane][k] = bidx[k]; }
  __syncthreads();

  if (lane < 16) {
    const float* LA = &mergeV[wave][lane][0];
    const int*   IA = &mergeI[wave][lane][0];
    const float* LB = &mergeV[wave][lane + 16][0];
    const int*   IB = &mergeI[wave][lane + 16][0];
    int sel[KNN];
    int p = 0, q = 0;
#pragma unroll
    for (int k = 0; k < KNN; ++k) {       // 2-pointer merge of two sorted lists
      const float va = (p < KNN) ? LA[p] : BIGF;
      const float vb = (q < KNN) ? LB[q] : BIGF;
      if (va <= vb) { sel[k] = IA[p]; ++p; }
      else          { sel[k] = IB[q]; ++q; }
    }

    // changingrate = sum_k min(|n x m_k|, |n (.) m_k|)
    const float nx = qp[3], ny = qp[4], nz = qp[5];
    float cr = 0.0f;
#pragma unroll
    for (int k = 0; k < KNN; ++k) {
      const float* mp = batch_pts + (size_t)sel[k] * 6 + 3;
      const float mx = mp[0], my = mp[1], mz = mp[2];
      const float cx = ny * mz - nz * my;
      const float cy = nz * mx - nx * mz;
      const float cz = nx * my - ny * mx;
      const float c1 = sqrtf(cx * cx + cy * cy + cz * cz);
      const float ex = nx * mx, ey = ny * my, ez = nz * mz;
      const float c2 = sqrtf(ex * ex + ey * ey + ez * ez);
      cr += fminf(c1, c2);
    }

    // Top-2 skeleton distances for this point (M=100, L2-resident).
    const float* sk = skel + (size_t)b * M_SKEL * 3;
    float d1 = BIGF, d2 = BIGF;
    for (int m = 0; m < M_SKEL; ++m) {
      const float dx = qx - sk[m * 3 + 0];
      const float dy = qy - sk[m * 3 + 1];
      const float dz = qz - sk[m * 3 + 2];
      const float d = dx * dx + dy * dy + dz * dz;
      if (d < d1)      { d2 = d1; d1 = d; }
      else if (d < d2) { d2 = d; }
    }
    atomicAdd(out, cr * fabsf(d1 - d2));   // loss_voronoi contribution
  }
}

// loss_chosen: for each skel point, distance to its nearest *other* skel point.
__global__ __launch_bounds__(128) void skel_chosen_kernel(const float* __restrict__ skel,
                                                          float* __restrict__ out) {
  const int b = blockIdx.x;
  const int t = threadIdx.x;
  if (t >= M_SKEL) return;
  const float* sk = skel + (size_t)b * M_SKEL * 3;
  const float sx = sk[t * 3 + 0], sy = sk[t * 3 + 1], sz = sk[t * 3 + 2];
  float bd = BIGF;
  int   bj = t;
  for (int j = 0; j < M_SKEL; ++j) {
    if (j == t) continue;                 // top-2 of d2 == self + nearest other
    const float dx = sx - sk[j * 3 + 0];
    const float dy = sy - sk[j * 3 + 1];
    const float dz = sz - sk[j * 3 + 2];
    const float d = dx * dx + dy * dy + dz * dz;
    if (d < bd) { bd = d; bj = j; }
  }
  const float dx = sx - sk[bj * 3 + 0];
  const float dy = sy - sk[bj * 3 + 1];
  const float dz = sz - sk[bj * 3 + 2];
  atomicAdd(out, -0.5f * sqrtf(dx * dx + dy * dy + dz * dz));
}

extern "C" void kernel_launch(void* const* d_in, const int* in_sizes, int n_in,
                              void* d_out, int out_size, void* d_ws, size_t ws_size,
                              hipStream_t stream) {
  (void)in_sizes; (void)n_in; (void)out_size; (void)d_ws; (void)ws_size;
  const float* xyz  = (const float*)d_in[0];   // (4, 4096, 6) f32
  // d_in[1] = num_class (scalar int) -> only feeds the dead loss_cd term
  const float* skel = (const float*)d_in[2];   // (4, 100, 3) f32
  float* out = (float*)d_out;                  // scalar f32

  zero_out_kernel<<<1, 1, 0, stream>>>(out);
  voronoi_kernel<<<128, 256, 0, stream>>>(xyz, skel, out);   // B*N/128 blocks
  skel_chosen_kernel<<<4, 128, 0, stream>>>(skel, out);
}